// MHA_76476187673063
// MI455X (gfx1250) — compile-verified
//
#include <hip/hip_runtime.h>
#include <math.h>

typedef __attribute__((ext_vector_type(2))) float v2f;
typedef __attribute__((ext_vector_type(8))) float v8f;

#define B_ 16
#define S_ 8192
#define E_ 1024
#define H_ 16
#define DK_ 64
#define SCALE 0.125f
#define SCHUNKS 8
#define SCHUNK_LEN (S_ / SCHUNKS)

// ---------------- small projection kernels ----------------

// qp[b,d] = bq[d] + sum_e q[b,e] * Wq[e,d]
__global__ void qproj_kernel(const float* __restrict__ q, const float* __restrict__ Wq,
                             const float* __restrict__ bq, float* __restrict__ qp) {
  int idx = blockIdx.x * blockDim.x + threadIdx.x;   // b*1024 + d
  int b = idx >> 10, d = idx & 1023;
  const float* qr = q + (size_t)b * E_;
  float acc = bq[d];
#pragma unroll 4
  for (int e = 0; e < E_; ++e) acc += qr[e] * Wq[(size_t)e * E_ + d];
  qp[idx] = acc;
}

// T[b,h,e] = SCALE * sum_{d<64} Wk[e, h*64+d] * qp[b, h*64+d]
__global__ void tvec_kernel(const float* __restrict__ Wk, const float* __restrict__ qp,
                            float* __restrict__ T) {
  int idx = blockIdx.x * blockDim.x + threadIdx.x;   // b*16384 + h*1024 + e
  int b = idx >> 14;
  int he = idx & 16383;
  int h = he >> 10, e = he & 1023;
  const float* wr = Wk + (size_t)e * E_ + h * DK_;
  const float* qr = qp + (size_t)b * E_ + h * DK_;
  float acc = 0.f;
#pragma unroll 8
  for (int d = 0; d < DK_; ++d) acc += wr[d] * qr[d];
  T[idx] = SCALE * acc;
}

// cbh[b,h] = SCALE * sum_{d<64} bk[h*64+d] * qp[b, h*64+d]
__global__ void cvec_kernel(const float* __restrict__ bk, const float* __restrict__ qp,
                            float* __restrict__ cbh) {
  int idx = threadIdx.x;  // 256 = b*16 + h
  int b = idx >> 4, h = idx & 15;
  const float* br = bk + h * DK_;
  const float* qr = qp + (size_t)b * E_ + h * DK_;
  float acc = 0.f;
#pragma unroll 8
  for (int d = 0; d < DK_; ++d) acc += br[d] * qr[d];
  cbh[idx] = SCALE * acc;
}

// ---------------- heavy pass 1: scores via WMMA f32 16x16x4 ----------------
// scores[b, s, h] = k[b,s,:] . T[b,h,:] + cbh[b,h]   (stored [B, S, H])
// A = k tile (16 s-rows x 4), B = T^T (4 x 16 heads)
__global__ void scores_kernel(const float* __restrict__ k, const float* __restrict__ T,
                              const float* __restrict__ cbh, float* __restrict__ scores) {
  int b    = blockIdx.y;
  int wave = threadIdx.x >> 5;
  int lane = threadIdx.x & 31;
  int m = lane & 15;   // A: s-row in tile; B: head column
  int p = lane >> 4;   // K-pair selector
  int s0 = (blockIdx.x * 4 + wave) * 16;

  const float* krow = k + ((size_t)(b * S_ + s0 + m)) * E_ + 2 * p;
  const float* Trow = T + ((size_t)(b * H_ + m)) * E_ + 2 * p;

  v8f acc = {0.f, 0.f, 0.f, 0.f, 0.f, 0.f, 0.f, 0.f};
#pragma unroll 4
  for (int kk = 0; kk < E_; kk += 4) {
    v2f a  = {krow[kk], krow[kk + 1]};
    v2f bb = {Trow[kk], Trow[kk + 1]};
    acc = __builtin_amdgcn_wmma_f32_16x16x4_f32(false, a, false, bb,
                                                (short)0, acc, false, false);
  }

  float cv = cbh[b * H_ + m];
  // C layout: vgpr j -> row (j + 8*p), col m
  float* srow = scores + ((size_t)(b * S_ + s0)) * H_ + m;
#pragma unroll
  for (int j = 0; j < 8; ++j)
    srow[(size_t)(j + 8 * p) * H_] = acc[j] + cv;
}

// ---------------- softmax over s for each (b,h), in place ----------------
__global__ void softmax_kernel(float* __restrict__ attn) {
  int bh = blockIdx.x;
  int b = bh >> 4, h = bh & 15;
  float* base = attn + (size_t)b * S_ * H_ + h;
  __shared__ float red[256];
  int t = threadIdx.x;

  float mx = -3.0e38f;
  for (int s = t; s < S_; s += 256) mx = fmaxf(mx, base[(size_t)s * H_]);
  red[t] = mx; __syncthreads();
  for (int off = 128; off; off >>= 1) {
    if (t < off) red[t] = fmaxf(red[t], red[t + off]);
    __syncthreads();
  }
  mx = red[0]; __syncthreads();

  float sum = 0.f;
  for (int s = t; s < S_; s += 256) sum += __expf(base[(size_t)s * H_] - mx);
  red[t] = sum; __syncthreads();
  for (int off = 128; off; off >>= 1) {
    if (t < off) red[t] += red[t + off];
    __syncthreads();
  }
  float inv = 1.0f / red[0];

  for (int s = t; s < S_; s += 256) {
    size_t i = (size_t)s * H_;
    base[i] = __expf(base[i] - mx) * inv;
  }
}

// attn_mean[b,s] = mean over h of attn[b,s,h]
__global__ void attnmean_kernel(const float* __restrict__ attn, float* __restrict__ om) {
  int idx = blockIdx.x * blockDim.x + threadIdx.x;  // b*S + s  (131072)
  const float4* r = (const float4*)(attn + (size_t)idx * H_);
  float4 a = r[0], bq4 = r[1], c4 = r[2], d4 = r[3];
  float s = a.x + a.y + a.z + a.w + bq4.x + bq4.y + bq4.z + bq4.w +
            c4.x + c4.y + c4.z + c4.w + d4.x + d4.y + d4.z + d4.w;
  om[idx] = s * (1.0f / 16.0f);
}

// ---------------- heavy pass 2: w = attn^T @ v via WMMA ----------------
// wpart[chunk][b,h,e] = sum_{s in chunk} attn[b,s,h] * v[b,s,e]
// A = attn^T (16 heads x 4 s), B = v (4 s x 16 e-cols)
__global__ void wacc_kernel(const float* __restrict__ attn, const float* __restrict__ v,
                            float* __restrict__ wpart) {
  int b     = blockIdx.y;
  int chunk = blockIdx.z;
  int wave  = threadIdx.x >> 5;
  int lane  = threadIdx.x & 31;
  int m = lane & 15;  // A: head row; B: e column
  int p = lane >> 4;
  int e0 = (blockIdx.x * 8 + wave) * 16;
  int sbase = chunk * SCHUNK_LEN;

  const float* arow = attn + ((size_t)(b * S_ + sbase + 2 * p)) * H_ + m;
  const float* vrow = v    + ((size_t)(b * S_ + sbase + 2 * p)) * E_ + e0 + m;

  v8f acc = {0.f, 0.f, 0.f, 0.f, 0.f, 0.f, 0.f, 0.f};
#pragma unroll 4
  for (int ss = 0; ss < SCHUNK_LEN; ss += 4) {
    v2f a  = {arow[(size_t)ss * H_], arow[(size_t)(ss + 1) * H_]};
    v2f bb = {vrow[(size_t)ss * E_], vrow[(size_t)(ss + 1) * E_]};
    acc = __builtin_amdgcn_wmma_f32_16x16x4_f32(false, a, false, bb,
                                                (short)0, acc, false, false);
  }

  // C: vgpr j -> row h = j + 8*p, col e0 + m
  float* wr = wpart + ((size_t)(chunk * B_ + b)) * H_ * E_;
#pragma unroll
  for (int j = 0; j < 8; ++j)
    wr[(size_t)(j + 8 * p) * E_ + e0 + m] = acc[j];
}

// wsum[i] = sum_chunk wpart[chunk][i]   (deterministic reduce, no FP atomics)
__global__ void wreduce_kernel(const float* __restrict__ wpart, float* __restrict__ wsum) {
  int i = blockIdx.x * blockDim.x + threadIdx.x;   // 262144
  float s = 0.f;
#pragma unroll
  for (int c = 0; c < SCHUNKS; ++c) s += wpart[(size_t)c * (B_ * H_ * E_) + i];
  wsum[i] = s;
}

// out[b, h*64+j] = bv[h*64+j] + sum_e wsum[b,h,e] * Wv[e, h*64+j]
__global__ void outproj_kernel(const float* __restrict__ wsum, const float* __restrict__ Wv,
                               const float* __restrict__ bv, float* __restrict__ out) {
  int bh = blockIdx.x;             // b*16 + h
  int b = bh >> 4, h = bh & 15;
  int j = threadIdx.x;             // 0..63
  const float* wr = wsum + (size_t)bh * E_;
  int col = h * DK_ + j;
  float acc = bv[col];
#pragma unroll 4
  for (int e = 0; e < E_; ++e) acc += wr[e] * Wv[(size_t)e * E_ + col];
  out[b * E_ + col] = acc;
}

// ---------------- launch ----------------
extern "C" void kernel_launch(void* const* d_in, const int* in_sizes, int n_in,
                              void* d_out, int out_size, void* d_ws, size_t ws_size,
                              hipStream_t stream) {
  const float* q  = (const float*)d_in[0];
  const float* k  = (const float*)d_in[1];
  const float* v  = (const float*)d_in[2];
  const float* Wq = (const float*)d_in[3];
  const float* bq = (const float*)d_in[4];
  const float* Wk = (const float*)d_in[5];
  const float* bk = (const float*)d_in[6];
  const float* Wv = (const float*)d_in[7];
  const float* bv = (const float*)d_in[8];
  float* out = (float*)d_out;            // [16*1024] out  ++  [16*8192] attn_mean

  float* ws = (float*)d_ws;
  float* qp    = ws;                                   // 16*1024       = 16384
  float* T     = qp + 16384;                           // 16*16*1024    = 262144
  float* cbh   = T + 262144;                           // 256
  float* attn  = cbh + 256;                            // 16*8192*16    = 2097152 (scores -> attn in place)
  float* wpart = attn + 2097152;                       // 8*16*16*1024  = 2097152
  float* wsum  = wpart + 2097152;                      // 262144

  qproj_kernel<<<64, 256, 0, stream>>>(q, Wq, bq, qp);
  tvec_kernel<<<1024, 256, 0, stream>>>(Wk, qp, T);
  cvec_kernel<<<1, 256, 0, stream>>>(bk, qp, cbh);

  scores_kernel<<<dim3(S_ / 64, B_), 128, 0, stream>>>(k, T, cbh, attn);
  softmax_kernel<<<B_ * H_, 256, 0, stream>>>(attn);
  attnmean_kernel<<<512, 256, 0, stream>>>(attn, out + B_ * E_);

  wacc_kernel<<<dim3(8, B_, SCHUNKS), 256, 0, stream>>>(attn, v, wpart);
  wreduce_kernel<<<1024, 256, 0, stream>>>(wpart, wsum);
  outproj_kernel<<<B_ * H_, DK_, 0, stream>>>(wsum, Wv, bv, out);
}